// GraphNeuralNetwork_20538533609732
// MI455X (gfx1250) — compile-verified
//
#include <hip/hip_runtime.h>

typedef __attribute__((ext_vector_type(2))) float v2f;
typedef __attribute__((ext_vector_type(8))) float v8f;

__global__ void degree_kernel(const int* __restrict__ src, const int* __restrict__ dst,
                              float* __restrict__ dego, float* __restrict__ degi, int E) {
    int e = blockIdx.x * blockDim.x + threadIdx.x;
    if (e >= E) return;
    atomicAdd(&dego[src[e]], 1.0f);
    atomicAdd(&degi[dst[e]], 1.0f);
}

__global__ void norm_kernel(float* __restrict__ dego, float* __restrict__ degi, int n) {
    int i = blockIdx.x * blockDim.x + threadIdx.x;
    if (i >= n) return;
    float d0 = dego[i];
    dego[i] = d0 > 0.f ? __frsqrt_rn(fmaxf(d0, 1.f)) : 0.f;
    float d1 = degi[i];
    degi[i] = d1 > 0.f ? __frsqrt_rn(fmaxf(d1, 1.f)) : 0.f;
}

// y[node][0..F) = x[node][0..F) * s[node]
template<int F>
__global__ void scale_rows_kernel(const float* __restrict__ x, const float* __restrict__ s,
                                  float* __restrict__ y, int n) {
    int i = blockIdx.x * blockDim.x + threadIdx.x;
    if (i >= n) return;
    float sv = s[i];
    const float* xr = x + (long long)i * F;
    float* yr = y + (long long)i * F;
#pragma unroll
    for (int f = 0; f < F; ++f) yr[f] = xr[f] * sv;
}

// Zero-padded copies of W [K x C] -> Wp [KP x CP], b [C] -> bp [CP].
__global__ void pad_weights_kernel(const float* __restrict__ W, const float* __restrict__ b,
                                   float* __restrict__ Wp, float* __restrict__ bp,
                                   int K, int C, int KP, int CP) {
    int i = blockIdx.x * blockDim.x + threadIdx.x;
    int total = KP * CP + CP;
    if (i >= total) return;
    if (i < KP * CP) {
        int k = i / CP, c = i % CP;
        Wp[i] = (k < K && c < C) ? W[k * C + c] : 0.f;
    } else {
        int c = i - KP * CP;
        bp[c] = (c < C) ? b[c] : 0.f;
    }
}

// Scatter-add messages: agg[dst][0..F) += feat[src][0..F).
// feat row stride IS, agg row stride OS.
template<int F, int IS, int OS>
__global__ void edge_aggregate_kernel(const float* __restrict__ feat, const int* __restrict__ src,
                                      const int* __restrict__ dst, float* __restrict__ agg, int E) {
    int e = blockIdx.x * blockDim.x + threadIdx.x;
    if (e >= E) return;
    int s = src[e], d = dst[e];
    const float* fs = feat + (long long)s * IS;
    float* ad = agg + (long long)d * OS;
#pragma unroll
    for (int f = 0; f < F; ++f) atomicAdd(&ad[f], fs[f]);
}

// D[16 nodes x COUT] = (scale_in * in[16 x KP]) @ Wp[KP x CP] + bp, optional ReLU
// and per-node scale_out. One wave32 per 16-node tile; V_WMMA_F32_16X16X4_F32,
// K in chunks of 4, columns tiled by 16. All fragment loads unconditional
// (in stride ISTRIDE >= KP zero-padded, Wp/bp pre-padded). When OSTRIDE == CP,
// all stores are unconditional too: padding columns get explicit zeros.
template<int KP, int CP, int COUT, int ISTRIDE, int OSTRIDE,
         bool RELU, bool SCALE_IN, bool SCALE_OUT>
__global__ void node_transform_wmma(const float* __restrict__ in,
                                    const float* __restrict__ innorm,
                                    const float* __restrict__ outnorm,
                                    const float* __restrict__ Wp,
                                    const float* __restrict__ bp,
                                    float* __restrict__ out, int n) {
    constexpr int KB = KP / 4;
    constexpr int CT = CP / 16;
    int tiles = (n + 15) >> 4;
    int tile = blockIdx.x * (blockDim.x >> 5) + (threadIdx.x >> 5);
    if (tile >= tiles) return;            // wave-uniform: EXEC stays all-ones
    int base = tile << 4;
    int lane = threadIdx.x & 31;
    int half = lane >> 4;                 // 0: lanes 0-15, 1: lanes 16-31
    int mrow = lane & 15;
    int nodeA = min(base + mrow, n - 1);
    float innv = 1.0f;
    if constexpr (SCALE_IN) innv = innorm[nodeA];
    const float* arow = in + (long long)nodeA * ISTRIDE;

    v8f acc[CT] = {};

#pragma unroll
    for (int kb = 0; kb < KB; ++kb) {
        // A 16x4 f32: VGPR0 = K=(kb*4 + half*2), VGPR1 = K+1, row M = lane&15
        int k0 = kb * 4 + half * 2;
        v2f a = *(const v2f*)(arow + k0);      // 8B-aligned b64 load
        a.x *= innv; a.y *= innv;
#pragma unroll
        for (int ct = 0; ct < CT; ++ct) {
            // B 4x16 f32: VGPR0 rows K0 (lanes 0-15)/K2 (16-31), VGPR1 rows K1/K3
            int col = ct * 16 + mrow;
            v2f bb;
            bb.x = Wp[ k0      * CP + col];
            bb.y = Wp[(k0 + 1) * CP + col];
            acc[ct] = __builtin_amdgcn_wmma_f32_16x16x4_f32(
                false, a, false, bb, (short)0, acc[ct], false, false);
        }
    }

    // D: VGPR r holds row M=r (lanes 0-15) and M=r+8 (lanes 16-31), N=lane&15
    int rbase = base + half * 8;
    if (base + 16 <= n) {
        // Full tile (always taken when n % 16 == 0): straight-line epilogue.
#pragma unroll
        for (int ct = 0; ct < CT; ++ct) {
            int col = ct * 16 + mrow;
            float bv = bp[col];
#pragma unroll
            for (int r = 0; r < 8; ++r) {
                int node = rbase + r;
                float v = acc[ct][r] + bv;
                if constexpr (RELU) v = fmaxf(v, 0.f);
                if constexpr (SCALE_OUT) v *= outnorm[node];
                if constexpr (OSTRIDE == CP) {
                    // unconditional store; pad columns get zeros (cndmask, no branch)
                    out[(long long)node * OSTRIDE + col] = (col < COUT) ? v : 0.f;
                } else {
                    if (col < COUT) out[(long long)node * OSTRIDE + col] = v;
                }
            }
        }
    } else {
        // Generic tail tile.
#pragma unroll
        for (int ct = 0; ct < CT; ++ct) {
            int col = ct * 16 + mrow;
            float bv = bp[col];
#pragma unroll
            for (int r = 0; r < 8; ++r) {
                int node = rbase + r;
                if (node < n) {
                    float v = acc[ct][r] + bv;
                    if constexpr (RELU) v = fmaxf(v, 0.f);
                    if constexpr (SCALE_OUT) v *= outnorm[node];
                    if (col < COUT) {
                        out[(long long)node * OSTRIDE + col] = v;
                    } else if (col < OSTRIDE) {
                        out[(long long)node * OSTRIDE + col] = 0.f;
                    }
                }
            }
        }
    }
}

extern "C" void kernel_launch(void* const* d_in, const int* in_sizes, int n_in,
                              void* d_out, int out_size, void* d_ws, size_t ws_size,
                              hipStream_t stream) {
    const float* x   = (const float*)d_in[0];
    const int*   ei  = (const int*)d_in[1];
    const float* W1  = (const float*)d_in[2];
    const float* b1  = (const float*)d_in[3];
    const float* W2  = (const float*)d_in[4];
    const float* b2  = (const float*)d_in[5];
    const float* Wfc = (const float*)d_in[6];
    const float* bfc = (const float*)d_in[7];
    float* outp = (float*)d_out;

    const int n = in_sizes[0] / 10;     // 100000 nodes
    const int E = in_sizes[1] / 2;      // 1.6M edges
    const int* src = ei;
    const int* dst = ei + E;

    float* ws   = (float*)d_ws;
    float* dego = ws;                    // becomes out_norm in place
    float* degi = ws + (long long)n;     // becomes in_norm in place
    float* bufA = ws + 2LL  * n;         // xs [stride 10, 10n]; later h2 [stride 32, 32n]
    float* bufB = ws + 34LL * n;         // agg1 [stride 12, 12n]; later agg2 [stride 20, 20n]
    float* bufC = ws + 54LL * n;         // h1 * out_norm [stride 32, 32n]
    float* wpad = ws + 86LL * n;         // padded weights/biases
    float* W1p = wpad;            float* b1p = W1p + 12 * 32;   // 384 + 32
    float* W2p = b1p + 32;        float* b2p = W2p + 20 * 32;   // 640 + 32
    float* Wfp = b2p + 32;        float* bfp = Wfp + 32 * 16;   // 512 + 16

    const int tb = 256;
    hipMemsetAsync(dego, 0, sizeof(float) * 2 * (size_t)n, stream);
    degree_kernel<<<(E + tb - 1) / tb, tb, 0, stream>>>(src, dst, dego, degi, E);
    norm_kernel<<<(n + tb - 1) / tb, tb, 0, stream>>>(dego, degi, n);

    pad_weights_kernel<<<(12 * 32 + 32 + tb - 1) / tb, tb, 0, stream>>>(W1, b1, W1p, b1p, 10, 20, 12, 32);
    pad_weights_kernel<<<(20 * 32 + 32 + tb - 1) / tb, tb, 0, stream>>>(W2, b2, W2p, b2p, 20, 30, 20, 32);
    pad_weights_kernel<<<(32 * 16 + 16 + tb - 1) / tb, tb, 0, stream>>>(Wfc, bfc, Wfp, bfp, 30, 10, 32, 16);

    // h = x * out_norm (pre-scaled layer-1 messages)
    scale_rows_kernel<10><<<(n + tb - 1) / tb, tb, 0, stream>>>(x, dego, bufA, n);

    hipMemsetAsync(bufB, 0, sizeof(float) * 12 * (size_t)n, stream);
    edge_aggregate_kernel<10, 10, 12><<<(E + tb - 1) / tb, tb, 0, stream>>>(bufA, src, dst, bufB, E);

    const int tiles   = (n + 15) / 16;
    const int tblocks = (tiles + 7) / 8;   // 8 waves (tiles) per 256-thread block
    // h1s = relu((in_norm*agg1)@W1 + b1) * out_norm, stride 32 (cols 20-31 zeroed)
    node_transform_wmma<12, 32, 20, 12, 32, true, true, true>
        <<<tblocks, 256, 0, stream>>>(bufB, degi, dego, W1p, b1p, bufC, n);

    hipMemsetAsync(bufB, 0, sizeof(float) * 20 * (size_t)n, stream);
    edge_aggregate_kernel<20, 32, 20><<<(E + tb - 1) / tb, tb, 0, stream>>>(bufC, src, dst, bufB, E);

    // h2 = relu((in_norm*agg2)@W2 + b2), stride 32 (cols 30-31 zeroed)
    node_transform_wmma<20, 32, 30, 20, 32, true, true, false>
        <<<tblocks, 256, 0, stream>>>(bufB, degi, dego, W2p, b2p, bufA, n);

    // out = h2 @ Wfc + bfc (packed N x 10)
    node_transform_wmma<32, 16, 10, 32, 10, false, false, false>
        <<<tblocks, 256, 0, stream>>>(bufA, degi, dego, Wfp, bfp, outp, n);
}